// TnoCausalV12_20014547599491
// MI455X (gfx1250) — compile-verified
//
#include <hip/hip_runtime.h>
#include <math.h>

// ---------------------------------------------------------------------------
// Causal long conv via FFT (fft_size = 16384 = 16*16*16*4) + exact GELU.
// Radix-16 DIF/inverse stages run on the fp32 WMMA pipe
// (v_wmma_f32_16x16x4_f32): each batch of 16 butterflies is a 16x16 complex
// matrix product = 16 WMMA instructions.  Forward is DIF (natural -> digit-
// reversed); the pointwise multiply happens in digit-reversed order (the
// filter FFT uses the identical DIF, so orders match); the inverse runs the
// exact stage composition in reverse with conjugated twiddles/matrices.
// Twiddles use one sincos + complex-rotation recurrences (trans-op friendly).
// Transpose staging uses CDNA5 async global->LDS copies.
// ---------------------------------------------------------------------------

typedef float v2f __attribute__((ext_vector_type(2)));
typedef float v8f __attribute__((ext_vector_type(8)));

#define PI_F 3.14159265358979323846f

#define B_SZ   8
#define N_SEQ  8192
#define D_CH   1024
#define FFT_N  16384

__device__ __forceinline__ float2 cmul(float2 a, float2 b) {
  return make_float2(a.x * b.x - a.y * b.y, a.x * b.y + a.y * b.x);
}
__device__ __forceinline__ float2 csqr(float2 a) {
  return make_float2(a.x * a.x - a.y * a.y, 2.0f * a.x * a.y);
}

// Build per-lane A-matrix fragments of the 16x16 DFT matrix for
// V_WMMA_F32_16X16X4_F32.  A is 16(M=row r) x 4(K=col s) per instruction:
// lanes 0-15 hold M=lane, VGPR0=K0, VGPR1=K1; lanes 16-31 hold K2,K3.
// DFT16[r][s] = exp(-2*pi*i*r*s/16) = cos - i*sin.
__device__ __forceinline__ void build_dft16_frags(int lane, v2f* aC, v2f* aS, v2f* aSN) {
  const int m   = lane & 15;
  const int khi = (lane >> 4) << 1;           // 0 or 2
#pragma unroll
  for (int q = 0; q < 4; ++q) {
    const int s0 = (q << 2) + khi;
    const float a0 = (2.0f * PI_F / 16.0f) * (float)(m * s0);
    const float a1 = (2.0f * PI_F / 16.0f) * (float)(m * (s0 + 1));
    float c0, c1, sn0, sn1;
    __sincosf(a0, &sn0, &c0);
    __sincosf(a1, &sn1, &c1);
    aC[q][0]  =  c0;  aC[q][1]  =  c1;   // cos
    aS[q][0]  =  sn0; aS[q][1]  =  sn1;  // +sin
    aSN[q][0] = -sn0; aSN[q][1] = -sn1;  // -sin
  }
}

// One radix-16 stage over the whole 16384-point LDS array.
// LOG2JC: jc = Ns/16 stride (10, 6, 2).  INV=false: DIF stage (DFT16 matrix,
// post-twiddle exp(-2pi i j r / Ns)).  INV=true: exact inverse of that stage
// (pre-twiddle exp(+2pi i j s / Ns), conjugate DFT16 matrix; the 1/R scales
// are folded into the pointwise multiply).
template <int LOG2JC, bool INV>
__device__ __forceinline__ void r16_stage(float2* lds,
                                          const v2f* aC, const v2f* aS, const v2f* aSN,
                                          int wave, int lane) {
  const int   jc      = 1 << LOG2JC;
  const float tw_unit = 2.0f * PI_F / (float)(jc << 4);   // 2*pi/Ns
  const int   ncol    = lane & 15;
  const int   shi     = (lane >> 4) << 1;                 // K half: 0 or 2
  const int   rbase   = (lane >> 4) << 3;                 // D rows: 0 or 8

  for (int batch = wave; batch < 64; batch += 8) {        // 1024 butterflies total
    const int g    = (batch << 4) + ncol;                 // this column's butterfly
    const int blk  = g >> LOG2JC;
    const int j    = g & (jc - 1);
    const int base = (blk << (LOG2JC + 4)) + j;

    // Gather B fragments (4x16 per instruction): lane supplies K = s0, s0+1.
    v2f br[4], bi[4];
    if (INV) {
      // Pre-twiddle with conj twiddle exp(+i * 2pi * j * s / Ns); twiddles for
      // s = shi, shi+1, shi+4, ... generated by rotation from one sincos.
      float sw, cw;
      __sincosf(tw_unit * (float)j, &sw, &cw);
      const float2 w1 = make_float2(cw, sw);
      const float2 w2 = csqr(w1);
      const float2 w4 = csqr(w2);
      float2 cur = (shi == 0) ? make_float2(1.0f, 0.0f) : w2;   // w^shi
#pragma unroll
      for (int q = 0; q < 4; ++q) {
        const int s0 = (q << 2) + shi;
        const float2 u0 = cmul(lds[base + (s0)     * jc], cur);
        const float2 u1 = cmul(lds[base + (s0 + 1) * jc], cmul(cur, w1));
        br[q][0] = u0.x; br[q][1] = u1.x;
        bi[q][0] = u0.y; bi[q][1] = u1.y;
        cur = cmul(cur, w4);
      }
    } else {
#pragma unroll
      for (int q = 0; q < 4; ++q) {
        const int s0 = (q << 2) + shi;
        const float2 u0 = lds[base + (s0)     * jc];
        const float2 u1 = lds[base + (s0 + 1) * jc];
        br[q][0] = u0.x; br[q][1] = u1.x;
        bi[q][0] = u0.y; bi[q][1] = u1.y;
      }
    }

    // Complex 16x16 x 16x16 product on the fp32 WMMA pipe (16 instructions).
    v8f dr = {}; v8f di = {};
    if (!INV) {
      // Vr = cos*Ur + sin*Ui ; Vi = cos*Ui - sin*Ur
#pragma unroll
      for (int q = 0; q < 4; ++q)
        dr = __builtin_amdgcn_wmma_f32_16x16x4_f32(false, aC[q],  false, br[q], (short)0, dr, false, false);
#pragma unroll
      for (int q = 0; q < 4; ++q)
        dr = __builtin_amdgcn_wmma_f32_16x16x4_f32(false, aS[q],  false, bi[q], (short)0, dr, false, false);
#pragma unroll
      for (int q = 0; q < 4; ++q)
        di = __builtin_amdgcn_wmma_f32_16x16x4_f32(false, aC[q],  false, bi[q], (short)0, di, false, false);
#pragma unroll
      for (int q = 0; q < 4; ++q)
        di = __builtin_amdgcn_wmma_f32_16x16x4_f32(false, aSN[q], false, br[q], (short)0, di, false, false);
    } else {
      // conj matrix: Vr = cos*Ur - sin*Ui ; Vi = cos*Ui + sin*Ur
#pragma unroll
      for (int q = 0; q < 4; ++q)
        dr = __builtin_amdgcn_wmma_f32_16x16x4_f32(false, aC[q],  false, br[q], (short)0, dr, false, false);
#pragma unroll
      for (int q = 0; q < 4; ++q)
        dr = __builtin_amdgcn_wmma_f32_16x16x4_f32(false, aSN[q], false, bi[q], (short)0, dr, false, false);
#pragma unroll
      for (int q = 0; q < 4; ++q)
        di = __builtin_amdgcn_wmma_f32_16x16x4_f32(false, aC[q],  false, bi[q], (short)0, di, false, false);
#pragma unroll
      for (int q = 0; q < 4; ++q)
        di = __builtin_amdgcn_wmma_f32_16x16x4_f32(false, aS[q],  false, br[q], (short)0, di, false, false);
    }

    // Scatter D (16x16, 8 rows per lane half); forward applies post-twiddle
    // exp(-i*2pi*j*r/Ns) via rotation recurrence (1 sincos per batch).
    if (!INV) {
      float sw, cw;
      __sincosf(-tw_unit * (float)j, &sw, &cw);
      const float2 w1 = make_float2(cw, sw);
      float2 cur;
      if (rbase == 0) {
        cur = make_float2(1.0f, 0.0f);
      } else {
        cur = csqr(csqr(csqr(w1)));                        // w^8
      }
#pragma unroll
      for (int v = 0; v < 8; ++v) {
        lds[base + (rbase + v) * jc] = cmul(make_float2(dr[v], di[v]), cur);
        cur = cmul(cur, w1);
      }
    } else {
#pragma unroll
      for (int v = 0; v < 8; ++v)
        lds[base + (rbase + v) * jc] = make_float2(dr[v], di[v]);
    }
  }
}

// Final radix-4 stage (Ns=4, j=0 -> no twiddles; pure VALU).
template <bool INV>
__device__ __forceinline__ void r4_stage(float2* lds, int tid) {
  for (int g = tid; g < FFT_N / 4; g += 256) {
    const int b = g << 2;
    const float2 u0 = lds[b], u1 = lds[b + 1], u2 = lds[b + 2], u3 = lds[b + 3];
    const float2 e0 = make_float2(u0.x + u2.x, u0.y + u2.y);
    const float2 e1 = make_float2(u0.x - u2.x, u0.y - u2.y);
    const float2 o0 = make_float2(u1.x + u3.x, u1.y + u3.y);
    const float2 o1 = make_float2(u1.x - u3.x, u1.y - u3.y);
    float2 v1, v3;
    if (!INV) {  // -i * o1
      v1 = make_float2(e1.x + o1.y, e1.y - o1.x);
      v3 = make_float2(e1.x - o1.y, e1.y + o1.x);
    } else {     // +i * o1
      v1 = make_float2(e1.x - o1.y, e1.y + o1.x);
      v3 = make_float2(e1.x + o1.y, e1.y - o1.x);
    }
    lds[b]     = make_float2(e0.x + o0.x, e0.y + o0.y);
    lds[b + 1] = v1;
    lds[b + 2] = make_float2(e0.x - o0.x, e0.y - o0.y);
    lds[b + 3] = v3;
  }
}

__device__ __forceinline__ void fft_fwd(float2* lds, const v2f* aC, const v2f* aS,
                                        const v2f* aSN, int wave, int lane, int tid) {
  r16_stage<10, false>(lds, aC, aS, aSN, wave, lane); __syncthreads();
  r16_stage<6,  false>(lds, aC, aS, aSN, wave, lane); __syncthreads();
  r16_stage<2,  false>(lds, aC, aS, aSN, wave, lane); __syncthreads();
  r4_stage<false>(lds, tid);                          __syncthreads();
}

__device__ __forceinline__ void fft_inv(float2* lds, const v2f* aC, const v2f* aS,
                                        const v2f* aSN, int wave, int lane, int tid) {
  r4_stage<true>(lds, tid);                           __syncthreads();
  r16_stage<2,  true>(lds, aC, aS, aSN, wave, lane);  __syncthreads();
  r16_stage<6,  true>(lds, aC, aS, aSN, wave, lane);  __syncthreads();
  r16_stage<10, true>(lds, aC, aS, aSN, wave, lane);  __syncthreads();
}

// ---------------------------------------------------------------------------
// Tiled 32x32 transpose: in (R x C) -> out (C x R), batched over blockIdx.z.
// Tile fill uses CDNA5 async global->LDS B128 copies (bypasses VGPRs,
// tracked with ASYNCcnt).
// ---------------------------------------------------------------------------
__global__ __launch_bounds__(256) void transpose_k(const float* __restrict__ in,
                                                   float* __restrict__ out,
                                                   int R, int C) {
  __shared__ float tile[32][33];
  const size_t bo = (size_t)blockIdx.z * (size_t)R * (size_t)C;
  const int r0 = blockIdx.x * 32, c0 = blockIdx.y * 32;
  const int tid = threadIdx.x;

  // Each thread async-copies one 16-byte segment: 32 rows x (8 x 16B).
  {
    const int row = tid >> 3;                    // 0..31
    const int seg = tid & 7;                     // 0..7
    const float* gsrc = in + bo + (size_t)(r0 + row) * C + (c0 + (seg << 2));
    const unsigned int       ldsoff = (unsigned int)(uintptr_t)&tile[row][seg << 2];
    const unsigned long long gaddr  = (unsigned long long)(uintptr_t)gsrc;
    asm volatile("global_load_async_to_lds_b128 %0, %1, off"
                 :: "v"(ldsoff), "v"(gaddr)
                 : "memory");
    asm volatile("s_wait_asynccnt 0x0" ::: "memory");
  }
  __syncthreads();

  const int tx = tid & 31, ty = tid >> 5;
#pragma unroll
  for (int i = 0; i < 32; i += 8)
    out[bo + (size_t)(c0 + ty + i) * R + (r0 + tx)] = tile[tx][ty + i];
}

// ---------------------------------------------------------------------------
// Filter FFT: one workgroup per channel; tt (d,n) -> Kf (d, 16384 complex),
// stored in the DIF digit-reversed order.
// ---------------------------------------------------------------------------
__global__ __launch_bounds__(256) void filter_fft_kernel(const float* __restrict__ tt,
                                                         float2* __restrict__ Kf) {
  extern __shared__ float2 lds[];
  const int tid = threadIdx.x, lane = tid & 31, wave = tid >> 5;
  v2f aC[4], aS[4], aSN[4];
  build_dft16_frags(lane, aC, aS, aSN);

  const float* src = tt + (size_t)blockIdx.x * N_SEQ;
  for (int i = tid; i < N_SEQ; i += 256)          lds[i] = make_float2(src[i], 0.0f);
  for (int i = N_SEQ + tid; i < FFT_N; i += 256)  lds[i] = make_float2(0.0f, 0.0f);
  __syncthreads();

  fft_fwd(lds, aC, aS, aSN, wave, lane, tid);

  float2* dst = Kf + (size_t)blockIdx.x * FFT_N;
  for (int i = tid; i < FFT_N; i += 256) dst[i] = lds[i];
}

// ---------------------------------------------------------------------------
// Fused conv: one workgroup per (b,c) sequence.  FFT -> pointwise * Kf (in
// permuted order, with 1/N folded in) -> inverse FFT -> exact GELU, in-place
// on the transposed staging buffer.
// ---------------------------------------------------------------------------
__global__ __launch_bounds__(256) void fftconv_kernel(const float2* __restrict__ Kf,
                                                      float* __restrict__ xt) {
  extern __shared__ float2 lds[];
  const int tid = threadIdx.x, lane = tid & 31, wave = tid >> 5;
  v2f aC[4], aS[4], aSN[4];
  build_dft16_frags(lane, aC, aS, aSN);

  const int seq = blockIdx.x;              // seq = b*1024 + c
  const int c   = seq & (D_CH - 1);
  float* xs = xt + (size_t)seq * N_SEQ;

  // Prefetch this channel's 128KB spectrum (L2-resident) so it streams in
  // behind the forward FFT.  128B per prefetch, 4 per thread.
  const float2* kf = Kf + (size_t)c * FFT_N;
  for (int i = (tid << 4); i < FFT_N; i += (256 << 4))
    __builtin_prefetch((const void*)(kf + i), 0, 0);

  for (int i = tid; i < N_SEQ; i += 256)          lds[i] = make_float2(xs[i], 0.0f);
  for (int i = N_SEQ + tid; i < FFT_N; i += 256)  lds[i] = make_float2(0.0f, 0.0f);
  __syncthreads();

  fft_fwd(lds, aC, aS, aSN, wave, lane, tid);

  const float scale = 1.0f / (float)FFT_N;
  for (int i = tid; i < FFT_N; i += 256) {
    const float2 a = lds[i];
    const float2 b = kf[i];
    lds[i] = make_float2((a.x * b.x - a.y * b.y) * scale,
                         (a.x * b.y + a.y * b.x) * scale);
  }
  __syncthreads();

  fft_inv(lds, aC, aS, aSN, wave, lane, tid);

  for (int i = tid; i < N_SEQ; i += 256) {
    const float y = lds[i].x;
    xs[i] = 0.5f * y * (1.0f + erff(y * 0.70710678118654752f));  // exact GELU
  }
}

// ---------------------------------------------------------------------------
extern "C" void kernel_launch(void* const* d_in, const int* in_sizes, int n_in,
                              void* d_out, int out_size, void* d_ws, size_t ws_size,
                              hipStream_t stream) {
  const float* x = (const float*)d_in[0];   // (b, n, d) f32
  const float* t = (const float*)d_in[1];   // (n, d)    f32
  float* out = (float*)d_out;               // (b, n, d) f32

  // Workspace layout: xt (b,d,n) 256MB | tt (d,n) 32MB | Kf (d,16384 cplx) 134MB
  float*  xt = (float*)d_ws;
  float*  tt = xt + (size_t)B_SZ * D_CH * N_SEQ;
  float2* Kf = (float2*)(tt + (size_t)D_CH * N_SEQ);

  const dim3 blk(256);
  const size_t fft_lds = (size_t)FFT_N * sizeof(float2);  // 128 KB dynamic LDS

  // 1. Stage transposed copies so each sequence is contiguous.
  transpose_k<<<dim3(N_SEQ / 32, D_CH / 32, B_SZ), blk, 0, stream>>>(x, xt, N_SEQ, D_CH);
  transpose_k<<<dim3(N_SEQ / 32, D_CH / 32, 1),    blk, 0, stream>>>(t, tt, N_SEQ, D_CH);

  // 2. Filter FFTs (1024 sequences).
  filter_fft_kernel<<<dim3(D_CH), blk, fft_lds, stream>>>(tt, Kf);

  // 3. Fused FFT-conv + GELU (8192 sequences), in-place on xt.
  fftconv_kernel<<<dim3(B_SZ * D_CH), blk, fft_lds, stream>>>(Kf, xt);

  // 4. Transpose back to (b, n, d).
  transpose_k<<<dim3(D_CH / 32, N_SEQ / 32, B_SZ), blk, 0, stream>>>(xt, out, D_CH, N_SEQ);
}